// VectorQuantizerRR_21371757265036
// MI455X (gfx1250) — compile-verified
//
#include <hip/hip_runtime.h>
#include <hip/hip_bf16.h>

typedef __attribute__((ext_vector_type(2))) float v2f;
typedef __attribute__((ext_vector_type(8))) float v8f;

#define KC    8192        // codebook size
#define DIM   256         // embedding dim
#define NTOK  16384       // 16*32*32 rows
#define DECAY 0.99f
#define BETA  0.25f
#define EPSV  1e-5f

#define BROW   260        // padded LDS row stride (floats)
#define TILEF  (16 * BROW)        // floats per B buffer (4160)
#define TILEB  (TILEF * 4)        // bytes per B buffer (16640, 16B aligned)

// ---------------------------------------------------------------------------
__global__ __launch_bounds__(256) void k_zero(float* __restrict__ p, int n) {
    int i = blockIdx.x * 256 + threadIdx.x;
    if (i < n) p[i] = 0.0f;
}

// ---------------------------------------------------------------------------
// e_sq[k] = sum_d w[k,d]^2
__global__ __launch_bounds__(256) void k_esq(const float* __restrict__ w, float* __restrict__ esq) {
    __shared__ float red[256];
    int k = blockIdx.x;
    float v = w[(size_t)k * DIM + threadIdx.x];
    red[threadIdx.x] = v * v;
    __syncthreads();
    for (int s = 128; s > 0; s >>= 1) {
        if (threadIdx.x < s) red[threadIdx.x] += red[threadIdx.x + s];
        __syncthreads();
    }
    if (threadIdx.x == 0) esq[k] = red[0];
}

// ---------------------------------------------------------------------------
// Distance GEMM + argmin. f32 WMMA (exact vs fp32 reference), A resident in
// registers for the whole K sweep, B tiles double-buffered in LDS via
// GLOBAL_LOAD_ASYNC_TO_LDS_B128 (ASYNCcnt), 4 independent WMMA accumulators.
__global__ __launch_bounds__(128)
void vq_dist_argmin(const float* __restrict__ z_e, const float* __restrict__ weight,
                    const float* __restrict__ esq, int* __restrict__ indices) {
    __shared__ __align__(16) float Bs[2][TILEF];   // 2 x 16 rows x 260 floats

    const int tid  = threadIdx.x;
    const int lane = tid & 31;
    const int wave = tid >> 5;
    const int n_wave = blockIdx.x * 64 + wave * 16;

    // flat[n, d] = z_e[(n>>10)*D*1024 + d*1024 + (n&1023)]
    const int bimg = n_wave >> 10;
    const int hw   = n_wave & 1023;
    const float* zbase = z_e + (size_t)bimg * (DIM * 1024) + hw;

    const int m    = lane & 15;           // A row fed by this lane
    const int dpar = (lane >> 4) * 2;     // lanes 16-31 carry k=2,3 slice
    const int col  = lane & 15;           // C/D column held by this lane

    // --- A into registers, WMMA 16x4-f32 layout, full D=256 (128 VGPRs) ---
    float areg[128];
#pragma unroll
    for (int s = 0; s < 64; ++s) {
        const int d0 = s * 4 + dpar;
        areg[2 * s]     = zbase[(size_t)d0 * 1024 + m];
        areg[2 * s + 1] = zbase[(size_t)(d0 + 1) * 1024 + m];
    }

    // LDS byte offset of Bs (LDS aperture: low 32 bits of flat address)
    const unsigned lds_base = (unsigned)(uintptr_t)&Bs[0][0];
    // each wave async-copies rows [wave*4, wave*4+4) of the 16-row tile:
    // 8 x b128 per wave, lane-striped 16B chunks.
    const unsigned lane16 = lane * 4;     // float offset within half-row

#define ISSUE_TILE(KB, BUF)                                                          \
    do {                                                                             \
        _Pragma("unroll")                                                            \
        for (int j = 0; j < 8; ++j) {                                                \
            const int r  = wave * 4 + (j >> 1);                                      \
            const int cc = (j & 1) * 128 + lane16;                                   \
            unsigned la  = lds_base + (unsigned)(BUF) * TILEB                        \
                           + (unsigned)(r * BROW + cc) * 4u;                         \
            unsigned long long ga =                                                  \
                (unsigned long long)(weight + (size_t)((KB) + r) * DIM + cc);        \
            asm volatile("global_load_async_to_lds_b128 %0, %1, off"                 \
                         :: "v"(la), "v"(ga) : "memory");                            \
        }                                                                            \
    } while (0)

    float best[8];
    int   bestk[8];
#pragma unroll
    for (int i = 0; i < 8; ++i) { best[i] = 3.0e38f; bestk[i] = 0; }

    ISSUE_TILE(0, 0);                                 // prologue: tile 0 -> buf 0

    for (int kb = 0, it = 0; kb < KC; kb += 16, ++it) {
        const int buf = it & 1;
        __syncthreads();                              // readers of buf^1 are done
        if (kb + 16 < KC) {
            ISSUE_TILE(kb + 16, buf ^ 1);             // overlap next copy w/ compute
            asm volatile("s_wait_asynccnt 0x8" ::: "memory");  // my 8 current-buf copies done
        } else {
            asm volatile("s_wait_asynccnt 0x0" ::: "memory");
        }
        __syncthreads();                              // whole tile visible

        // --- 16x16 cross tile: 64 f32 WMMAs, 4 independent accumulators ---
        const float* bsrc = &Bs[buf][0];
        v8f c0 = {}, c1 = {}, c2 = {}, c3 = {};
#pragma unroll
        for (int s = 0; s < 64; s += 4) {
            v2f a0 = { areg[2 * s + 0], areg[2 * s + 1] };
            v2f a1 = { areg[2 * s + 2], areg[2 * s + 3] };
            v2f a2 = { areg[2 * s + 4], areg[2 * s + 5] };
            v2f a3 = { areg[2 * s + 6], areg[2 * s + 7] };
            const v2f b0 = *(const v2f*)&bsrc[col * BROW + (s + 0) * 4 + dpar];
            const v2f b1 = *(const v2f*)&bsrc[col * BROW + (s + 1) * 4 + dpar];
            const v2f b2 = *(const v2f*)&bsrc[col * BROW + (s + 2) * 4 + dpar];
            const v2f b3 = *(const v2f*)&bsrc[col * BROW + (s + 3) * 4 + dpar];
            c0 = __builtin_amdgcn_wmma_f32_16x16x4_f32(false, a0, false, b0, (short)0, c0, false, false);
            c1 = __builtin_amdgcn_wmma_f32_16x16x4_f32(false, a1, false, b1, (short)0, c1, false, false);
            c2 = __builtin_amdgcn_wmma_f32_16x16x4_f32(false, a2, false, b2, (short)0, c2, false, false);
            c3 = __builtin_amdgcn_wmma_f32_16x16x4_f32(false, a3, false, b3, (short)0, c3, false, false);
        }
        v8f c = (c0 + c1) + (c2 + c3);

        // dist = e_sq[k] - 2*cross   (row-constant x_sq irrelevant to argmin)
        const float e2 = esq[kb + col];
#pragma unroll
        for (int i = 0; i < 8; ++i) {
            float dv = e2 - 2.0f * c[i];
            if (dv < best[i]) { best[i] = dv; bestk[i] = kb + col; }
        }
    }
#undef ISSUE_TILE

    // cross-lane arg-reduction within each 16-lane half
#pragma unroll
    for (int i = 0; i < 8; ++i) {
        float v = best[i];
        int   k = bestk[i];
#pragma unroll
        for (int off = 8; off >= 1; off >>= 1) {
            float ov = __shfl_xor(v, off, 32);
            int   ok = __shfl_xor(k, off, 32);
            if (ov < v || (ov == v && ok < k)) { v = ov; k = ok; }
        }
        if ((lane & 15) == 0) {
            int row = i + ((lane >> 4) << 3);   // VGPR i: lanes0-15 -> row i, 16-31 -> i+8
            indices[n_wave + row] = k;
        }
    }
}

// ---------------------------------------------------------------------------
__global__ __launch_bounds__(256)
void k_scatter(const float* __restrict__ z_e, const int* __restrict__ indices,
               float* __restrict__ embed_sum, float* __restrict__ counts) {
    int n = blockIdx.x * 256 + threadIdx.x;
    int d = blockIdx.y;
    int idx = indices[n];
    float v = z_e[(size_t)(n >> 10) * (DIM * 1024) + (size_t)d * 1024 + (n & 1023)];
    atomicAdd(&embed_sum[(size_t)idx * DIM + d], v);
    if (d == 0) atomicAdd(&counts[idx], 1.0f);
}

// ---------------------------------------------------------------------------
__global__ __launch_bounds__(256)
void k_ema_cluster(const float* __restrict__ cs_in, const float* __restrict__ counts,
                   float* __restrict__ nc_out, float* __restrict__ usage,
                   float* __restrict__ n_sum) {
    int k = blockIdx.x * 256 + threadIdx.x;
    float cnt = counts[k];
    float nc = cs_in[k] * DECAY + cnt * (1.0f - DECAY);
    nc_out[k] = nc;
    usage[k]  = cnt * (1.0f / (float)NTOK);
    atomicAdd(n_sum, nc);
}

__global__ __launch_bounds__(256)
void k_ema_weight(const float* __restrict__ ea, const float* __restrict__ esum,
                  const float* __restrict__ nc, const float* __restrict__ n_sum,
                  float* __restrict__ navg, float* __restrict__ nw) {
    int k = blockIdx.x;
    int d = threadIdx.x;
    size_t i = (size_t)k * DIM + d;
    float na = ea[i] * DECAY + esum[i] * (1.0f - DECAY);
    navg[i] = na;
    float n  = *n_sum;
    float c  = nc[k];
    float cs = (c + EPSV) / (n + (float)KC * EPSV) * n;
    nw[i] = na / cs;
}

// ---------------------------------------------------------------------------
__global__ __launch_bounds__(256)
void k_quant(const float* __restrict__ z_e, const int* __restrict__ indices,
             const float* __restrict__ nw, float* __restrict__ zq,
             float* __restrict__ loss_sum) {
    __shared__ float red[8];
    int n = blockIdx.x * 256 + threadIdx.x;
    int d = blockIdx.y;
    int idx = indices[n];
    float q = nw[(size_t)idx * DIM + d];
    size_t off = (size_t)(n >> 10) * (DIM * 1024) + (size_t)d * 1024 + (n & 1023);
    float z = z_e[off];
    zq[off] = q;
    float diff = z - q;
    float s = diff * diff;
    for (int o = 16; o > 0; o >>= 1) s += __shfl_down(s, o, 32);
    int lane = threadIdx.x & 31, wv = threadIdx.x >> 5;
    if (lane == 0) red[wv] = s;
    __syncthreads();
    if (threadIdx.x == 0) {
        float t = 0.0f;
        for (int i = 0; i < 8; ++i) t += red[i];
        atomicAdd(loss_sum, t);
    }
}

__global__ void k_loss(const float* __restrict__ loss_sum, float* __restrict__ out2) {
    float msd = *loss_sum * (1.0f / (float)(NTOK * DIM));
    out2[0] = BETA * msd;   // commitment_loss
    out2[1] = msd;          // codebook_loss
}

// ---------------------------------------------------------------------------
extern "C" void kernel_launch(void* const* d_in, const int* in_sizes, int n_in,
                              void* d_out, int out_size, void* d_ws, size_t ws_size,
                              hipStream_t stream) {
    const float* z_e          = (const float*)d_in[0];   // [16,256,32,32]
    const float* weight       = (const float*)d_in[1];   // [8192,256]
    const float* cluster_size = (const float*)d_in[2];   // [8192]
    const float* embed_avg    = (const float*)d_in[3];   // [8192,256]

    float* out = (float*)d_out;
    float* zq_out      = out;                       // 4,194,304
    float* loss_out    = out + 4194304;             // commit @+0, codebook @+1
    float* usage_out   = out + 4194306;             // 8192
    float* ncl_out     = out + 4202498;             // 8192
    float* navg_out    = out + 4210690;             // 2,097,152

    float* ws        = (float*)d_ws;
    float* esq       = ws;                           // 8192
    float* counts    = ws + 8192;                    // 8192
    float* n_sum     = ws + 16384;                   // 1
    float* loss_sum  = ws + 16385;                   // 1 (pad to 16640)
    float* embed_sum = ws + 16640;                   // 2,097,152
    float* new_w     = ws + 16640 + 2097152;         // 2,097,152
    int*   indices   = (int*)(ws + 16640 + 2 * 2097152);  // 16384 ints

    const int zero_n = (16640 - 8192) + KC * DIM;    // counts..embed_sum span
    k_zero<<<(zero_n + 255) / 256, 256, 0, stream>>>(counts, zero_n);

    k_esq<<<KC, 256, 0, stream>>>(weight, esq);

    vq_dist_argmin<<<NTOK / 64, 128, 0, stream>>>(z_e, weight, esq, indices);

    dim3 gsc(NTOK / 256, DIM);
    k_scatter<<<gsc, 256, 0, stream>>>(z_e, indices, embed_sum, counts);

    k_ema_cluster<<<KC / 256, 256, 0, stream>>>(cluster_size, counts, ncl_out, usage_out, n_sum);
    k_ema_weight<<<KC, 256, 0, stream>>>(embed_avg, embed_sum, ncl_out, n_sum, navg_out, new_w);

    dim3 gq(NTOK / 256, DIM);
    k_quant<<<gq, 256, 0, stream>>>(z_e, indices, new_w, zq_out, loss_sum);
    k_loss<<<1, 1, 0, stream>>>(loss_sum, loss_out);
}